// AttentionDecoder_36867999269342
// MI455X (gfx1250) — compile-verified
//
#include <hip/hip_runtime.h>
#include <hip/hip_bf16.h>
#include <math.h>

#define Bsz 64
#define Ssz 256
#define Isz 512
#define Hsz 512
#define Csz 100
#define Esz 128
#define Tsz 32
#define NSTEP 31
#define XDIM (Isz + Esz)   // 640
#define GDIM (4 * Hsz)     // 2048
#define NEG_INF -1e30f

typedef __attribute__((ext_vector_type(2))) float v2f;
typedef __attribute__((ext_vector_type(8))) float v8f;

// ---------------------------------------------------------------------------
// Async-stage 16 rows x K floats from global into LDS (row stride strideP
// floats, padded to avoid bank conflicts). Uses CDNA5 GLOBAL_LOAD_ASYNC_TO_LDS
// (ASYNCcnt-tracked DMA); caller must s_wait_asynccnt + __syncthreads().
// ---------------------------------------------------------------------------
__device__ __forceinline__ void stage_rows_async(const float* __restrict__ g0,
                                                 int lda, int K, int strideP,
                                                 float* s) {
    int segs = K >> 2;               // float4 chunks per row
    int nchunk = 16 * segs;
    for (int q = threadIdx.x; q < nchunk; q += blockDim.x) {
        int row = q / segs;
        int seg = q - row * segs;
        const float* g = g0 + (size_t)row * lda + seg * 4;
        unsigned lds = (unsigned)(uintptr_t)(s + row * strideP + seg * 4);
        asm volatile("global_load_async_to_lds_b128 %0, %1, off"
                     :: "v"(lds), "v"(g) : "memory");
    }
}

__device__ __forceinline__ void wait_async_all() {
    asm volatile("s_wait_asynccnt 0x0" ::: "memory");
    __syncthreads();
}

// ---------------------------------------------------------------------------
// f32 16x16x4 WMMA fragment layout (ISA 7.12.2):
//   A 16x4: lanes 0-15 -> M rows, K pair {0,1} (half 0) / {2,3} (half 1)
//   B 4x16: lanes 0-15 -> N cols, same K-pair split by half-wave
//   C/D:    v8f, lane<16: D[v][lane], lane>=16: D[8+v][lane-16]
// A comes from LDS (staged), B (weights, row-major (N,K)) from global/L2.
// ---------------------------------------------------------------------------
__device__ __forceinline__ v8f wmma_k_lds(const float* __restrict__ sA, int sP,
                                          const float* __restrict__ W, int ldb,
                                          int l15, int half, int bcol, int K,
                                          v8f acc) {
    for (int k0 = 0; k0 < K; k0 += 4) {
        int ka = k0 + half * 2;
        v2f a, b;
        a.x = sA[l15 * sP + ka];
        a.y = sA[l15 * sP + ka + 1];
        b.x = W[(size_t)bcol * ldb + ka];
        b.y = W[(size_t)bcol * ldb + ka + 1];
        acc = __builtin_amdgcn_wmma_f32_16x16x4_f32(false, a, false, b,
                                                    (short)0, acc, false, false);
    }
    return acc;
}

// ---------------------------------------------------------------------------
// Generic GEMM: out[M,N] = A[M,K] * W[N,K]^T + bias[N]. One wave per 16x16
// tile, 8 waves/block sharing one A-row-tile staged in LDS (tiles_n % 8 == 0).
// Dynamic LDS: 16*(K+4)*4 bytes.
// ---------------------------------------------------------------------------
__global__ void k_gemm_bias(const float* __restrict__ A, int K, int lda,
                            const float* __restrict__ W,
                            const float* __restrict__ bias,
                            float* __restrict__ out, int N,
                            int tiles_total, int tiles_n) {
    extern __shared__ float sA[];                       // 16 x (K+4)
    int wave0 = blockIdx.x * (blockDim.x >> 5);
    int tile_m = wave0 / tiles_n;                       // uniform per block
    stage_rows_async(A + (size_t)tile_m * 16 * lda, lda, K, K + 4, sA);
    wait_async_all();

    int wave = wave0 + (threadIdx.x >> 5);
    if (wave >= tiles_total) return;
    int tile_n = wave % tiles_n;
    int lane = threadIdx.x & 31;
    int half = lane >> 4;
    int l15  = lane & 15;
    int bcol = tile_n * 16 + l15;
    v8f acc = {};
    acc = wmma_k_lds(sA, K + 4, W, K, l15, half, bcol, K, acc);
    int col = tile_n * 16 + l15;
    for (int v = 0; v < 8; v++) {
        int row = tile_m * 16 + v + half * 8;
        out[(size_t)row * N + col] = acc[v] + bias[col];
    }
}

// ---------------------------------------------------------------------------
// Gates GEMM: gates[64,2048] = x[64,640]*W_ih^T + h[64,512]*W_hh^T + b_ih+b_hh
// Both A operands staged via async DMA. Dynamic LDS: 16*(644+516)*4 bytes.
// ---------------------------------------------------------------------------
__global__ void k_gates(const float* __restrict__ x, const float* __restrict__ h,
                        const float* __restrict__ W_ih, const float* __restrict__ W_hh,
                        const float* __restrict__ b_ih, const float* __restrict__ b_hh,
                        float* __restrict__ gates) {
    extern __shared__ float smem[];
    float* sX = smem;                                   // 16 x (XDIM+4)
    float* sH = smem + 16 * (XDIM + 4);                 // 16 x (Hsz+4)
    const int tiles_n = GDIM / 16;                      // 128
    int wave0 = blockIdx.x * 8;
    int tile_m = wave0 / tiles_n;                       // uniform per block
    stage_rows_async(x + (size_t)tile_m * 16 * XDIM, XDIM, XDIM, XDIM + 4, sX);
    stage_rows_async(h + (size_t)tile_m * 16 * Hsz,  Hsz,  Hsz,  Hsz + 4,  sH);
    wait_async_all();

    int wave = wave0 + (threadIdx.x >> 5);
    int tile_n = wave % tiles_n;
    int lane = threadIdx.x & 31;
    int half = lane >> 4;
    int l15  = lane & 15;
    int bcol = tile_n * 16 + l15;
    v8f acc = {};
    acc = wmma_k_lds(sX, XDIM + 4, W_ih, XDIM, l15, half, bcol, XDIM, acc);
    acc = wmma_k_lds(sH, Hsz + 4,  W_hh, Hsz,  l15, half, bcol, Hsz,  acc);
    int col = tile_n * 16 + l15;
    for (int v = 0; v < 8; v++) {
        int row = tile_m * 16 + v + half * 8;
        gates[(size_t)row * GDIM + col] = acc[v] + b_ih[col] + b_hh[col];
    }
}

// ---------------------------------------------------------------------------
// Attention: one block per batch element b (256 threads).
// ---------------------------------------------------------------------------
__global__ void k_attend(const float* __restrict__ projH,
                         const float* __restrict__ pp,
                         const float* __restrict__ W_score,
                         const float* __restrict__ b_score,
                         const unsigned char* __restrict__ mask,
                         const float* __restrict__ batch_H,
                         const float* __restrict__ embed,
                         const int* __restrict__ text,
                         float* __restrict__ x, int t) {
    __shared__ float s_pp[Hsz];
    __shared__ float s_wsc[Hsz];
    __shared__ float s_alpha[Ssz];
    __shared__ float red[256];
    int b = blockIdx.x;
    int tid = threadIdx.x;

    s_pp[tid]        = pp[(size_t)b * Hsz + tid];
    s_pp[tid + 256]  = pp[(size_t)b * Hsz + tid + 256];
    s_wsc[tid]       = W_score[tid];
    s_wsc[tid + 256] = W_score[tid + 256];
    __syncthreads();

    // fused tanh-dot: one (b,s) per thread
    int s = tid;
    const float* rowp = projH + ((size_t)(b * Ssz + s)) * Hsz;
    float e = b_score[0];
    for (int hh = 0; hh < Hsz; hh += 4) {
        float4 v = *(const float4*)(rowp + hh);
        e += s_wsc[hh + 0] * tanhf(v.x + s_pp[hh + 0]);
        e += s_wsc[hh + 1] * tanhf(v.y + s_pp[hh + 1]);
        e += s_wsc[hh + 2] * tanhf(v.z + s_pp[hh + 2]);
        e += s_wsc[hh + 3] * tanhf(v.w + s_pp[hh + 3]);
    }
    if (!mask[(size_t)b * Ssz + s]) e = NEG_INF;

    // block softmax over s
    red[tid] = e;
    __syncthreads();
    for (int off = 128; off > 0; off >>= 1) {
        if (tid < off) red[tid] = fmaxf(red[tid], red[tid + off]);
        __syncthreads();
    }
    float mx = red[0];
    __syncthreads();
    float ex = expf(e - mx);
    red[tid] = ex;
    __syncthreads();
    for (int off = 128; off > 0; off >>= 1) {
        if (tid < off) red[tid] += red[tid + off];
        __syncthreads();
    }
    float inv = 1.0f / red[0];
    s_alpha[s] = ex * inv;
    __syncthreads();

    // context matvec: each thread owns 2 feature columns
    float c0 = 0.f, c1 = 0.f;
    for (int ss = 0; ss < Ssz; ss++) {
        float al = s_alpha[ss];
        const float* r = batch_H + ((size_t)(b * Ssz + ss)) * Isz;
        c0 += al * r[tid];
        c1 += al * r[tid + 256];
    }
    x[(size_t)b * XDIM + tid]       = c0;
    x[(size_t)b * XDIM + tid + 256] = c1;

    if (tid < Esz) {
        int ch = text[b * Tsz + t];
        x[(size_t)b * XDIM + Isz + tid] = embed[(size_t)ch * Esz + tid];
    }
}

// ---------------------------------------------------------------------------
// LSTM pointwise: updates h, c in place from gates.
// ---------------------------------------------------------------------------
__device__ __forceinline__ float sigm(float v) { return 1.0f / (1.0f + expf(-v)); }

__global__ void k_lstm(const float* __restrict__ gates,
                       float* __restrict__ h, float* __restrict__ c) {
    int idx = blockIdx.x * blockDim.x + threadIdx.x;   // 64*512
    if (idx >= Bsz * Hsz) return;
    int b = idx >> 9;
    int j = idx & (Hsz - 1);
    const float* g = gates + (size_t)b * GDIM;
    float ig = g[j];
    float fg = g[j + Hsz];
    float gg = g[j + 2 * Hsz];
    float og = g[j + 3 * Hsz];
    float cn = sigm(fg) * c[idx] + sigm(ig) * tanhf(gg);
    float hn = sigm(og) * tanhf(cn);
    c[idx] = cn;
    h[idx] = hn;
}

// ---------------------------------------------------------------------------
// Logits GEMM (N=100, guarded): out[b, t, :] = h[64,512]*W_gen[100,512]^T + b_gen
// ---------------------------------------------------------------------------
__global__ void k_logits(const float* __restrict__ h,
                         const float* __restrict__ W_gen,
                         const float* __restrict__ b_gen,
                         float* __restrict__ out, int t) {
    const int tiles_n = (Csz + 15) / 16;            // 7
    const int tiles_total = (Bsz / 16) * tiles_n;   // 28
    int wave = (blockIdx.x * blockDim.x + threadIdx.x) >> 5;
    if (wave >= tiles_total) return;
    int tile_m = wave / tiles_n;
    int tile_n = wave % tiles_n;
    int lane = threadIdx.x & 31;
    int half = lane >> 4;
    int l15  = lane & 15;
    int arow = tile_m * 16 + l15;
    int bcol = tile_n * 16 + l15;
    v8f acc = {};
    for (int k0 = 0; k0 < Hsz; k0 += 4) {
        int ka = k0 + half * 2;
        v2f a, bb;
        a.x = h[(size_t)arow * Hsz + ka];
        a.y = h[(size_t)arow * Hsz + ka + 1];
        bb.x = (bcol < Csz) ? W_gen[(size_t)bcol * Hsz + ka]     : 0.f;
        bb.y = (bcol < Csz) ? W_gen[(size_t)bcol * Hsz + ka + 1] : 0.f;
        acc = __builtin_amdgcn_wmma_f32_16x16x4_f32(false, a, false, bb,
                                                    (short)0, acc, false, false);
    }
    int col = tile_n * 16 + l15;
    if (col < Csz) {
        for (int v = 0; v < 8; v++) {
            int row = tile_m * 16 + v + half * 8;   // batch index
            out[((size_t)row * NSTEP + t) * Csz + col] = acc[v] + b_gen[col];
        }
    }
}

__global__ void k_zero(float* __restrict__ p, int n) {
    int i = blockIdx.x * blockDim.x + threadIdx.x;
    if (i < n) p[i] = 0.f;
}

__global__ void k_targets(const int* __restrict__ text, float* __restrict__ out) {
    int i = blockIdx.x * blockDim.x + threadIdx.x;   // 64*31
    if (i >= Bsz * NSTEP) return;
    int b = i / NSTEP;
    int tt = i % NSTEP;
    out[(size_t)Bsz * NSTEP * Csz + (size_t)b * NSTEP + tt] =
        (float)text[b * Tsz + tt + 1];
}

// ---------------------------------------------------------------------------
extern "C" void kernel_launch(void* const* d_in, const int* in_sizes, int n_in,
                              void* d_out, int out_size, void* d_ws, size_t ws_size,
                              hipStream_t stream) {
    const float* batch_H = (const float*)d_in[0];
    const int*   text    = (const int*)d_in[1];
    const unsigned char* mask = (const unsigned char*)d_in[2];
    const float* W_i2h   = (const float*)d_in[3];
    const float* b_i2h   = (const float*)d_in[4];
    const float* W_h2h   = (const float*)d_in[5];
    const float* b_h2h   = (const float*)d_in[6];
    const float* W_score = (const float*)d_in[7];
    const float* b_score = (const float*)d_in[8];
    const float* embed   = (const float*)d_in[9];
    const float* W_ih    = (const float*)d_in[10];
    const float* b_ih    = (const float*)d_in[11];
    const float* W_hh    = (const float*)d_in[12];
    const float* b_hh    = (const float*)d_in[13];
    const float* W_gen   = (const float*)d_in[14];
    const float* b_gen   = (const float*)d_in[15];
    float* out = (float*)d_out;

    float* ws = (float*)d_ws;
    const size_t OFF_PROJ = 0;                              // 16384*512
    const size_t OFF_H    = OFF_PROJ + (size_t)Bsz * Ssz * Hsz;
    const size_t OFF_C    = OFF_H + (size_t)Bsz * Hsz;
    const size_t OFF_PP   = OFF_C + (size_t)Bsz * Hsz;
    const size_t OFF_X    = OFF_PP + (size_t)Bsz * Hsz;
    const size_t OFF_G    = OFF_X + (size_t)Bsz * XDIM;
    float* projH = ws + OFF_PROJ;
    float* hbuf  = ws + OFF_H;
    float* cbuf  = ws + OFF_C;
    float* pp    = ws + OFF_PP;
    float* xbuf  = ws + OFF_X;
    float* gates = ws + OFF_G;

    const size_t SH_GEMM  = (size_t)16 * (Hsz + 4) * sizeof(float);      // 33 KB
    const size_t SH_GATES = (size_t)16 * ((XDIM + 4) + (Hsz + 4)) * sizeof(float); // 74 KB

    // zero h and c (contiguous)
    k_zero<<<(2 * Bsz * Hsz + 255) / 256, 256, 0, stream>>>(hbuf, 2 * Bsz * Hsz);

    // proj_H = batch_H @ W_i2h^T + b_i2h : (16384 x 512) tiles = 1024 x 32
    {
        int tiles_total = (Bsz * Ssz / 16) * (Hsz / 16);   // 32768 waves
        int blocks = (tiles_total + 7) / 8;                // 8 waves/block
        k_gemm_bias<<<blocks, 256, SH_GEMM, stream>>>(batch_H, Isz, Isz, W_i2h,
                                                      b_i2h, projH, Hsz,
                                                      tiles_total, Hsz / 16);
    }

    k_targets<<<(Bsz * NSTEP + 255) / 256, 256, 0, stream>>>(text, out);

    for (int t = 0; t < NSTEP; t++) {
        // proj_prev = h @ W_h2h^T + b_h2h : (64 x 512) = 128 tiles
        {
            int tiles_total = (Bsz / 16) * (Hsz / 16);
            int blocks = (tiles_total + 7) / 8;
            k_gemm_bias<<<blocks, 256, SH_GEMM, stream>>>(hbuf, Hsz, Hsz, W_h2h,
                                                          b_h2h, pp, Hsz,
                                                          tiles_total, Hsz / 16);
        }
        // attention + context + embed concat -> x
        k_attend<<<Bsz, 256, 0, stream>>>(projH, pp, W_score, b_score, mask,
                                          batch_H, embed, text, xbuf, t);
        // gates = x@W_ih^T + h@W_hh^T + biases : 512 tiles
        k_gates<<<64, 256, SH_GATES, stream>>>(xbuf, hbuf, W_ih, W_hh,
                                               b_ih, b_hh, gates);
        // LSTM pointwise, updates h/c in place
        k_lstm<<<(Bsz * Hsz + 255) / 256, 256, 0, stream>>>(gates, hbuf, cbuf);
        // logits = h @ W_gen^T + b_gen -> out[b, t, :]
        k_logits<<<4, 256, 0, stream>>>(hbuf, W_gen, b_gen, out, t);
    }
}